// KANN_4578435137547
// MI455X (gfx1250) — compile-verified
//
#include <hip/hip_runtime.h>
#include <hip/hip_bf16.h>
#include <stdint.h>

// Problem constants (from reference)
#define N_WIDTH     8
#define N_ORDER     2
#define N_ELEMENTS  1024
#define N_NODES     2049          // N_ELEMENTS*N_ORDER + 1
#define N_SAMPLES   4096

// Flat float offsets into d_out (tuple concatenated in return order)
#define O_T    0
#define O_DT   (N_SAMPLES * N_WIDTH)             // 32768
#define O_DDT  (2 * N_SAMPLES * N_WIDTH)         // 65536
#define O_PHI  (3 * N_SAMPLES * N_WIDTH)         // 98304
#define PHI_SZ ((size_t)N_SAMPLES * N_WIDTH * N_NODES)   // 67,141,632

typedef float v4f __attribute__((ext_vector_type(4)));
typedef int   v4i __attribute__((ext_vector_type(4)));

// ---------------------------------------------------------------------------
// Kernel 1: zero-fill the three dense phi arrays (806 MB total) using the
// CDNA5 async LDS->global store path (ASYNCcnt) when available.
// Each lane streams 16B/iteration from a shared 4KB zero block in LDS.
// ---------------------------------------------------------------------------
__global__ void kan_fill_zero(float* __restrict__ phi_base, size_t nquads)
{
    __shared__ v4i zbuf[256];
    const int t = threadIdx.x;
    zbuf[t] = (v4i){0, 0, 0, 0};
    __syncthreads();

    size_t idx    = (size_t)blockIdx.x * blockDim.x + t;
    size_t stride = (size_t)gridDim.x * blockDim.x;
    v4i* gbase = (v4i*)phi_base;

#if __has_builtin(__builtin_amdgcn_global_store_async_from_lds_b128)
    // LDS source address (32-bit LDS offset survives the uintptr truncation)
    auto lsrc = (__attribute__((address_space(3))) v4i*)(uintptr_t)&zbuf[t];
    for (size_t q = idx; q < nquads; q += stride) {
        auto gdst = (__attribute__((address_space(1))) v4i*)(uintptr_t)(gbase + q);
        // (global dst, lds src, imm offset, imm cpol)
        __builtin_amdgcn_global_store_async_from_lds_b128(gdst, lsrc, 0, 0);
    }
  #if __has_builtin(__builtin_amdgcn_s_wait_asynccnt)
    __builtin_amdgcn_s_wait_asynccnt(0);
  #else
    asm volatile("s_wait_asynccnt 0x0" ::: "memory");
  #endif
#else
    // Fallback: plain streaming 128-bit nontemporal stores
    v4i z = (v4i){0, 0, 0, 0};
    for (size_t q = idx; q < nquads; q += stride)
        __builtin_nontemporal_store(z, gbase + q);
#endif
}

// ---------------------------------------------------------------------------
// Kernel 2: per-sample closed-form basis evaluation + scatter + tiny dots.
// Order-2 Lagrange on nodes {-1,0,1}:
//   phi  = { x(x-1)/2,  1-x^2,  x(x+1)/2 }
//   phi' = { x-1/2,     -2x,    x+1/2 }          (times 1/dx, dx = 1/2048)
//   phi''= { 1, -2, 1 }                          (times 1/dx^2)
// ---------------------------------------------------------------------------
__global__ void kan_compute(const float* __restrict__ x,
                            const float* __restrict__ w,
                            float* __restrict__ out)
{
    int i = blockIdx.x * blockDim.x + threadIdx.x;
    if (i >= N_SAMPLES) return;

    float xs = 2048.0f * x[i];                 // (N_NODES-1) * (x - X_MIN)/(X_MAX - X_MIN)
    int el = (int)floorf(xs * 0.5f);
    el = el < 0 ? 0 : (el > (N_ELEMENTS - 1) ? (N_ELEMENTS - 1) : el);
    int nl = el * N_ORDER;                     // left node id
    float xt = xs - (float)nl - 1.0f;          // local coord in [-1,1]

    const float s1 = 2048.0f;
    const float s2 = 2048.0f * 2048.0f;

    float p0 = 0.5f * xt * (xt - 1.0f);
    float p1 = 1.0f - xt * xt;
    float p2 = 0.5f * xt * (xt + 1.0f);

    float d0 = (xt - 0.5f) * s1;
    float d1 = (-2.0f * xt) * s1;
    float d2 = (xt + 0.5f) * s1;

    float q0 = s2;
    float q1 = -2.0f * s2;
    float q2 = s2;

    float* t_out   = out + O_T;
    float* dt_out  = out + O_DT;
    float* ddt_out = out + O_DDT;
    float* phi     = out + O_PHI;
    float* dphi    = phi + PHI_SZ;
    float* ddphi   = phi + 2 * PHI_SZ;

#pragma unroll
    for (int k = 0; k < N_WIDTH; ++k) {
        const float* wr = w + (size_t)k * N_NODES + nl;
        float w0 = wr[0], w1 = wr[1], w2 = wr[2];

        t_out[i * N_WIDTH + k]   = p0 * w0 + p1 * w1 + p2 * w2;
        dt_out[i * N_WIDTH + k]  = d0 * w0 + d1 * w1 + d2 * w2;
        ddt_out[i * N_WIDTH + k] = q0 * w0 + q1 * w1 + q2 * w2;

        size_t base = ((size_t)i * N_WIDTH + k) * N_NODES + (size_t)nl;
        phi[base + 0]   = p0;  phi[base + 1]   = p1;  phi[base + 2]   = p2;
        dphi[base + 0]  = d0;  dphi[base + 1]  = d1;  dphi[base + 2]  = d2;
        ddphi[base + 0] = q0;  ddphi[base + 1] = q1;  ddphi[base + 2] = q2;
    }
}

// ---------------------------------------------------------------------------
extern "C" void kernel_launch(void* const* d_in, const int* in_sizes, int n_in,
                              void* d_out, int out_size, void* d_ws, size_t ws_size,
                              hipStream_t stream)
{
    const float* x = (const float*)d_in[0];   // [4096]
    const float* w = (const float*)d_in[1];   // [8, 2049]
    float* out = (float*)d_out;

    // Zero the three dense phi arrays: 3 * 67,141,632 floats = 50,356,224 float4s
    size_t nquads = (3 * PHI_SZ) / 4;
    kan_fill_zero<<<8192, 256, 0, stream>>>(out + O_PHI, nquads);

    // Compute + scatter (overwrites the 9 nonzero slots per (i,k))
    kan_compute<<<(N_SAMPLES + 255) / 256, 256, 0, stream>>>(x, w, out);
}